// MCPBRNN_Generic_PETconstraint_variable_Out_constantLoss_53695681134737
// MI455X (gfx1250) — compile-verified
//
#include <hip/hip_runtime.h>
#include <math.h>

// ---------------------------------------------------------------------------
// MCPBRNN PET-constraint cell, hidden_size==1, SEQ==1.
// Serial recurrence (after algebraic folding, exact for all c since u2>=0):
//   c_{t+1} = (c + u1) - ooh*c - min(ol1*c, u2) - ooh*c*tanh(kh*c + k0h)
// Critical cycle per step: FMA(arg) -> v_tanh_f32 -> FMA(c_next).
// All per-step outputs are elementwise in (c_t, u_t) -> parallel kernel.
// ---------------------------------------------------------------------------

#define SPIN_N      365
#define TRAIN_N     1000000
#define RED_BLOCKS  256
#define RED_THREADS 256
#define PF_DIST     2048   // prefetch distance in steps (16 KB ahead)

// gfx1250 hardware v_tanh_f32 — probe: unguarded builtin. If this fails to
// compile, revert to the v_exp_f32 + v_rcp_f32 fallback.
__device__ __forceinline__ float fast_tanh(float a) {
  return __builtin_amdgcn_tanhf(a);
}

// sigmoid(k*c + k0) with kh = 0.5*k, k0h = 0.5*k0 (logistic-as-tanh, same as XLA)
__device__ __forceinline__ float sigmoid_half(float c, float kh, float k0h) {
  return fmaf(fast_tanh(fmaf(c, kh, k0h)), 0.5f, 0.5f);
}

// ---------------------------------------------------------------------------
// Stage 1: deterministic partial sums of y_obs[SPIN:hi] (sum, sum of squares)
// ---------------------------------------------------------------------------
__global__ void mcpbrnn_reduce_kernel(const float* __restrict__ y, double* __restrict__ P, int hi) {
  __shared__ double sh_s[RED_THREADS];
  __shared__ double sh_q[RED_THREADS];
  const int tid    = threadIdx.x;
  const int stride = RED_BLOCKS * RED_THREADS;
  double s = 0.0, q = 0.0;
  for (int i = SPIN_N + blockIdx.x * RED_THREADS + tid; i < hi; i += stride) {
    double v = (double)y[i];
    s += v;
    q += v * v;
  }
  sh_s[tid] = s; sh_q[tid] = q;
  __syncthreads();
  for (int off = RED_THREADS >> 1; off > 0; off >>= 1) {
    if (tid < off) { sh_s[tid] += sh_s[tid + off]; sh_q[tid] += sh_q[tid + off]; }
    __syncthreads();
  }
  if (tid == 0) {
    P[blockIdx.x]              = sh_s[0];
    P[RED_BLOCKS + blockIdx.x] = sh_q[0];
  }
}

// ---------------------------------------------------------------------------
// Stage 2: fixed-order final sum -> unbiased std; precompute gate constants.
// consts[0]=obs_std  [1]=oo1  [2]=ol1  [3]=0.5*k  [4]=0.5*k0
// ---------------------------------------------------------------------------
__global__ void mcpbrnn_finalize_kernel(const double* __restrict__ P,
                                        float* __restrict__ consts,
                                        const float* __restrict__ c_mean,
                                        const float* __restrict__ c_std,
                                        const float* __restrict__ w_yom,
                                        const float* __restrict__ w_ylm,
                                        const float* __restrict__ w_yfm,
                                        const float* __restrict__ b0_yom,
                                        const float* __restrict__ wb1_yom,
                                        int hi) {
  if (threadIdx.x != 0 || blockIdx.x != 0) return;
  double s = 0.0, q = 0.0;
  for (int i = 0; i < RED_BLOCKS; ++i) { s += P[i]; q += P[RED_BLOCKS + i]; }
  double n   = (double)(hi - SPIN_N);
  double var = (q - s * s / n) / (n - 1.0);
  consts[0] = (float)sqrt(var > 0.0 ? var : 0.0);

  double eo = exp((double)w_yom[0]);
  double el = exp((double)w_ylm[0]);
  double ef = exp((double)w_yfm[0]);
  double den = eo + el + ef;
  consts[1] = (float)(eo / den);                       // oo1
  consts[2] = (float)(el / den);                       // ol1
  double k  = (double)wb1_yom[0] / (double)c_std[0];   // wb1/so
  double k0 = (double)b0_yom[0] - (double)c_mean[0] * k;
  consts[3] = (float)(0.5 * k);
  consts[4] = (float)(0.5 * k0);
}

// ---------------------------------------------------------------------------
// Stage 3: the serial scan. One lane walks the whole sequence.
// Dependent cycle: FMA(arg) -> v_tanh_f32 -> FMA(c_next). The side chain
//   prehc = ((c + u1) - ooh*c) - min(ol1*c, u2)
// is depth-3 from c and completes under the tanh latency. Pre-update c_t is
// streamed straight into the c_n region of d_out (offset B). Input latency is
// hidden with global_prefetch_b8 one cacheline per 8-step block, 16 KB ahead.
// ---------------------------------------------------------------------------
__global__ void mcpbrnn_scan_kernel(const float* __restrict__ x,
                                    const int* __restrict__ tlp,
                                    const float* __restrict__ consts,
                                    float* __restrict__ dout, int B) {
  if (threadIdx.x != 0 || blockIdx.x != 0) return;
  int tl = *tlp;
  if (tl < 0) tl = 0;
  if (tl > B) tl = B;

  const float oo1 = consts[1];
  const float ol1 = consts[2];
  const float kh  = consts[3];
  const float k0h = consts[4];
  const float ooh = 0.5f * oo1;

  const float2* __restrict__ xv = (const float2*)(x) + tl;   // x rows, 8B each
  float* __restrict__ cout = dout + (size_t)B + tl;
  const int N = B - tl;

  float c = 0.0f;
  int s = 0;
  for (; s + 8 <= N; s += 8) {
    __builtin_prefetch((const void*)(xv + (s + PF_DIST)), 0, 3);  // global_prefetch_b8
#pragma unroll
    for (int j = 0; j < 8; ++j) {
      const float2 u = xv[s + j];
      cout[s + j] = c;                          // pre-update state c_t
      const float arg = fmaf(c, kh, k0h);       // chain op 1
      const float hc  = ooh * c;                // side depth 1
      const float a1  = c + u.x;                // side depth 1
      const float lc  = fminf(ol1 * c, u.y);    // side depth 2 (olc*c)
      const float th  = fast_tanh(arg);         // chain op 2 (v_tanh_f32)
      const float prehc = (a1 - hc) - lc;       // side depth 3
      c = fmaf(th, -hc, prehc);                 // chain op 3: prehc - hc*th
    }
  }
  for (; s < N; ++s) {
    const float2 u = xv[s];
    cout[s] = c;
    const float arg = fmaf(c, kh, k0h);
    const float hc  = ooh * c;
    const float a1  = c + u.x;
    const float lc  = fminf(ol1 * c, u.y);
    const float th  = fast_tanh(arg);
    const float prehc = (a1 - hc) - lc;
    c = fmaf(th, -hc, prehc);
  }
}

// ---------------------------------------------------------------------------
// Stage 4: embarrassingly parallel per-step outputs. Thread b reads c_t from
// the c_n region, x[b], and fills all 13 output regions (coalesced stores).
// ---------------------------------------------------------------------------
__global__ void mcpbrnn_outputs_kernel(const float* __restrict__ x,
                                       const int* __restrict__ tlp,
                                       const float* __restrict__ consts,
                                       float* __restrict__ dout, int B) {
  const int b = blockIdx.x * blockDim.x + threadIdx.x;
  if (b >= B) return;
  const int tl = *tlp;
  const size_t Bs = (size_t)B;

  if (b < tl) {
    dout[b]           = 0.0f;   // h_n
    dout[Bs + b]      = 0.0f;   // c_n
    dout[2*Bs + b]    = 0.0f;   // l_n
    dout[3*Bs + b]    = 0.0f;   // lc_n
    dout[4*Bs + b]    = 0.0f;   // bp_n
    dout[5*Bs + b]    = 0.0f;   // Gate_ib
    dout[6*Bs + b]    = 0.0f;   // Gate_oo
    dout[7*Bs + b]    = 0.0f;   // Gate_ol
    dout[8*Bs + b]    = 0.0f;   // Gate_olc
    dout[9*Bs + b]    = 0.0f;   // Gate_f
    dout[10*Bs + 2*b]     = 0.0f;  // h_nout[:,0]
    dout[10*Bs + 2*b + 1] = 0.0f;  // h_nout[:,1]
    dout[12*Bs + b]   = 0.0f;   // obs_std
    return;
  }

  const float obs_std = consts[0];
  const float oo1 = consts[1];
  const float ol1 = consts[2];
  const float kh  = consts[3];
  const float k0h = consts[4];

  const float c  = dout[Bs + b];          // pre-update state written by scan
  const float u2 = x[2*b + 1];

  const float sg  = sigmoid_half(c, kh, k0h);
  const float oo  = oo1 * sg;
  const float olc = (c > 0.0f) ? fminf(ol1, u2 / c) : ol1;
  const float f   = 1.0f - oo - olc;
  const float hn  = oo * c;

  dout[b]           = hn;         // h_n = oo*c  (bp contribution is 0)
  dout[Bs + b]      = c;          // c_n (rewrite same value)
  dout[2*Bs + b]    = ol1 * c;    // l_n
  dout[3*Bs + b]    = olc * c;    // lc_n
  dout[4*Bs + b]    = 0.0f;       // bp_n
  dout[5*Bs + b]    = 0.0f;       // Gate_ib
  dout[6*Bs + b]    = oo;         // Gate_oo
  dout[7*Bs + b]    = ol1;        // Gate_ol
  dout[8*Bs + b]    = olc;        // Gate_olc
  dout[9*Bs + b]    = f;          // Gate_f
  dout[10*Bs + 2*b]     = hn;       // h_nout[:,0]
  dout[10*Bs + 2*b + 1] = obs_std;  // h_nout[:,1]
  dout[12*Bs + b]   = obs_std;    // obs_std
}

// ---------------------------------------------------------------------------
extern "C" void kernel_launch(void* const* d_in, const int* in_sizes, int n_in,
                              void* d_out, int out_size, void* d_ws, size_t ws_size,
                              hipStream_t stream) {
  const float* x      = (const float*)d_in[0];
  // d_in[1] = epoch (unused)
  const int*   tl     = (const int*)d_in[2];
  const float* y_obs  = (const float*)d_in[3];
  const float* c_mean = (const float*)d_in[4];
  const float* c_std  = (const float*)d_in[5];
  const float* w_yom  = (const float*)d_in[6];
  const float* w_ylm  = (const float*)d_in[7];
  const float* w_yfm  = (const float*)d_in[8];
  const float* b0     = (const float*)d_in[9];
  const float* wb1    = (const float*)d_in[10];
  float* out = (float*)d_out;

  const int B  = in_sizes[0] / 2;                 // x is [B,1,2]
  int hi = in_sizes[3];
  if (hi > TRAIN_N) hi = TRAIN_N;

  double* P      = (double*)d_ws;                 // 2*RED_BLOCKS doubles (4 KB)
  float*  consts = (float*)((char*)d_ws + 4096);  // 5 floats

  // 1) partial sums for obs_std (fully overwritten each call -> no init needed)
  mcpbrnn_reduce_kernel<<<RED_BLOCKS, RED_THREADS, 0, stream>>>(y_obs, P, hi);
  // 2) fixed-order finalize: obs_std + gate constants
  mcpbrnn_finalize_kernel<<<1, 32, 0, stream>>>(P, consts, c_mean, c_std,
                                                w_yom, w_ylm, w_yfm, b0, wb1, hi);
  // 3) serial scan (the true critical path)
  mcpbrnn_scan_kernel<<<1, 32, 0, stream>>>(x, tl, consts, out, B);
  // 4) parallel per-step outputs
  mcpbrnn_outputs_kernel<<<(B + 255) / 256, 256, 0, stream>>>(x, tl, consts, out, B);
}